// GIN_NodeWeightEncoder_2645699854452
// MI455X (gfx1250) — compile-verified
//
#include <hip/hip_runtime.h>
#include <hip/hip_bf16.h>

#define N_NODES 50000
#define N_EDGES 600000
#define DIM     128
#define BN_EPS  1e-5f

#define TILES_PER_BLOCK 5          // 3125 tiles = 625 blocks x 5
#define LDS_STRIDE      132        // 128 dwords + 4 pad dwords (TDM pad)

typedef __attribute__((ext_vector_type(2))) float v2f;
typedef __attribute__((ext_vector_type(8))) float v8f;
typedef unsigned int uint32x4 __attribute__((ext_vector_type(4)));
typedef int          int32x8  __attribute__((ext_vector_type(8)));
typedef int          int32x4  __attribute__((ext_vector_type(4)));

// ---------------------------------------------------------------------------
// Vectorized copy (float4 grid-stride)
// ---------------------------------------------------------------------------
__global__ void copy_vec4_kernel(const float4* __restrict__ in,
                                 float4* __restrict__ out, int n4) {
    int i = blockIdx.x * blockDim.x + threadIdx.x;
    int stride = gridDim.x * blockDim.x;
    for (; i < n4; i += stride) out[i] = in[i];
}

// ---------------------------------------------------------------------------
// Edge aggregation: h[dst] += x[src]   (wave-per-edge, float4 per lane)
// ---------------------------------------------------------------------------
__global__ void edge_aggregate_kernel(const float* __restrict__ x,
                                      const long long* __restrict__ src,
                                      const long long* __restrict__ dst,
                                      float* __restrict__ h) {
    int gid  = blockIdx.x * blockDim.x + threadIdx.x;
    int e    = gid >> 5;
    int lane = gid & 31;
    if (e >= N_EDGES) return;
    long long s = src[e];
    long long d = dst[e];
    const float4 v = *(const float4*)(x + s * DIM + lane * 4);
    float* dp = h + d * DIM + lane * 4;
    atomicAdd(dp + 0, v.x);
    atomicAdd(dp + 1, v.y);
    atomicAdd(dp + 2, v.z);
    atomicAdd(dp + 3, v.w);
}

// ---------------------------------------------------------------------------
// TDM: issue tensor_load_to_lds of one 16x128 fp32 tile (rows of A) into LDS.
// D# built per CDNA5 ISA §8.3/§8.4:
//  group0: count=1 | lds_addr | global_addr[56:0] | type=2
//  group1: data_size=4B, pad_enable (interval=128 dwords -> enc 6,
//          amount=4 dwords -> enc 3), tensor_dim0=128, tensor_dim1=N,
//          tile_dim0=128, tile_dim1=16, tensor_dim0_stride=128
// Padding makes the LDS row stride 132 dwords (bank-conflict avoidance).
// ---------------------------------------------------------------------------
__device__ __forceinline__ void tdm_issue_tile(const float* __restrict__ A,
                                               int tile, unsigned ldsByteOff) {
    unsigned long long gaddr =
        (unsigned long long)(uintptr_t)(A + (unsigned long long)tile * 16 * DIM);

    uint32x4 g0;
    g0[0] = 1u;                                   // count=1, flags/gather = 0
    g0[1] = ldsByteOff;                           // lds_addr (bytes)
    g0[2] = (unsigned)(gaddr & 0xFFFFFFFFull);    // global_addr[31:0]
    g0[3] = (unsigned)((gaddr >> 32) & 0x1FFFFFFull) | (2u << 30); // [56:32]|type=2

    int32x8 g1;
    g1[0] = (2 << 16)        // data_size = 4 bytes
          | (1 << 20)        // pad_enable
          | (6 << 22)        // pad_interval: 128 dwords
          | (3 << 25);       // pad_amount:   4 dwords
    g1[1] = (DIM & 0xFFFF) << 16;                 // tensor_dim0[15:0] = 128
    g1[2] = 0 | ((N_NODES & 0xFFFF) << 16);       // dim0 hi=0, tensor_dim1 lo16
    g1[3] = (N_NODES >> 16) | (DIM << 16);        // tensor_dim1 hi16, tile_dim0=128
    g1[4] = 16;                                   // tile_dim1=16, tile_dim2=0
    g1[5] = DIM;                                  // tensor_dim0_stride lo32 = 128
    g1[6] = 0;                                    // stride hi / dim1_stride lo
    g1[7] = 0;

    int32x4 gz4 = {0, 0, 0, 0};                   // groups 2/3 unused (2D)
    int32x8 gz8 = {0, 0, 0, 0, 0, 0, 0, 0};       // extra group (6-arg builtin)
    __builtin_amdgcn_tensor_load_to_lds(g0, g1, gz4, gz4, gz8, 0);
}

// ---------------------------------------------------------------------------
// GEMM: out = relu(A[N x 128] @ W[128 x 128] + bias), fp32 WMMA 16x16x4.
// 625 blocks x 5 row-tiles.  Per block: all 32 B fragments preloaded once
// (64 VGPRs/lane), A tiles streamed in by TDM with double-buffered LDS,
// pipelined with s_wait_tensorcnt.
// ---------------------------------------------------------------------------
__global__ __launch_bounds__(256)
void gemm128_relu_wmma_kernel(const float* __restrict__ A,
                              const float* __restrict__ W,
                              const float* __restrict__ bias,
                              float* __restrict__ out) {
    __shared__ float As[2][16 * LDS_STRIDE];   // only LDS object -> offset 0

    const int tid     = threadIdx.x;
    const int wave    = tid >> 5;
    const int lane    = tid & 31;
    const int half    = lane >> 4;   // lane half: K pair (A/B), row half (C)
    const int ln      = lane & 15;   // M (A) / N (B,C)
    const int colBase = wave * 16;

    // Preload all B fragments of this wave's 16 columns (W is L2-resident)
    v2f bfrag[32];
#pragma unroll
    for (int j = 0; j < 32; ++j) {
        int k = 4 * j + 2 * half;
        bfrag[j][0] = W[(k + 0) * DIM + colBase + ln];
        bfrag[j][1] = W[(k + 1) * DIM + colBase + ln];
    }
    const float bcol = bias[colBase + ln];

    const int tile0 = blockIdx.x * TILES_PER_BLOCK;

    if (wave == 0) tdm_issue_tile(A, tile0, 0u);

    for (int t = 0; t < TILES_PER_BLOCK; ++t) {
        const int buf = t & 1;
        if (wave == 0) {
            if (t + 1 < TILES_PER_BLOCK) {
                tdm_issue_tile(A, tile0 + t + 1,
                               (unsigned)(((t + 1) & 1) * 16 * LDS_STRIDE * 4));
                __builtin_amdgcn_s_wait_tensorcnt(1);  // tile t landed
            } else {
                __builtin_amdgcn_s_wait_tensorcnt(0);  // last tile landed
            }
        }
        __syncthreads();

        const float* at = &As[buf][0];
        // TDM writes LDS behind the compiler's back: escape the tile pointer
        // with a memory clobber so the A-fragment ds_loads are not folded to
        // undef / hoisted above the tensorcnt wait + barrier.
        asm volatile("" :: "v"(at) : "memory");

        v8f acc = {};
#pragma unroll
        for (int j = 0; j < 32; ++j) {
            const int k = 4 * j;
            v2f a;
            // A 16x4 fp32 fragment: VGPR0 = even K of pair, VGPR1 = odd K;
            // lanes 16-31 hold K pair {2,3}.
            a[0] = at[ln * LDS_STRIDE + k + 2 * half + 0];
            a[1] = at[ln * LDS_STRIDE + k + 2 * half + 1];
            acc = __builtin_amdgcn_wmma_f32_16x16x4_f32(
                /*neg_a=*/false, a, /*neg_b=*/false, bfrag[j],
                /*c_mod=*/(short)0, acc, /*reuse=*/false, false);
        }

        const int rowBase = (tile0 + t) * 16;
#pragma unroll
        for (int r = 0; r < 8; ++r) {
            // C layout: VGPR r -> row r (lanes 0-15) / row r+8 (lanes 16-31)
            int row = rowBase + half * 8 + r;
            float v = acc[r] + bcol;
            v = v > 0.0f ? v : 0.0f;
            out[(long long)row * DIM + colBase + ln] = v;
        }
        __syncthreads();   // buffer reuse fence before next TDM target
    }
}

// ---------------------------------------------------------------------------
// Zero a small float buffer
// ---------------------------------------------------------------------------
__global__ void zero_kernel(float* __restrict__ p, int n) {
    int i = blockIdx.x * blockDim.x + threadIdx.x;
    if (i < n) p[i] = 0.0f;
}

// ---------------------------------------------------------------------------
// BatchNorm stats over 128 channels: stats[c] = sum, stats[128+c] = sumsq
// ---------------------------------------------------------------------------
__global__ void bn_reduce128_kernel(const float* __restrict__ h,
                                    float* __restrict__ stats) {
    const int c    = threadIdx.x & 127;
    const int rgrp = threadIdx.x >> 7;      // 0 or 1
    float s = 0.0f, q = 0.0f;
    for (int row = blockIdx.x * 2 + rgrp; row < N_NODES; row += gridDim.x * 2) {
        float v = h[(long long)row * DIM + c];
        s += v;
        q += v * v;
    }
    __shared__ float ls[2][128];
    __shared__ float lq[2][128];
    ls[rgrp][c] = s;
    lq[rgrp][c] = q;
    __syncthreads();
    if (rgrp == 0) {
        atomicAdd(&stats[c],       ls[0][c] + ls[1][c]);
        atomicAdd(&stats[128 + c], lq[0][c] + lq[1][c]);
    }
}

__global__ void bn_apply128_kernel(const float* __restrict__ in,
                                   const float* __restrict__ stats,
                                   const float* __restrict__ gamma,
                                   const float* __restrict__ beta,
                                   float* __restrict__ out) {
    const float invN = 1.0f / (float)N_NODES;
    int i = blockIdx.x * blockDim.x + threadIdx.x;
    const int total = N_NODES * DIM;
    const int stride = gridDim.x * blockDim.x;
    for (; i < total; i += stride) {
        int c = i & 127;
        float mean = stats[c] * invN;
        float var  = stats[128 + c] * invN - mean * mean;
        float rs   = __frsqrt_rn(var + BN_EPS);
        out[i] = (in[i] - mean) * rs * gamma[c] + beta[c];
    }
}

// ---------------------------------------------------------------------------
// Final projection: out2 = relu(H[N x 128] @ W[128 x 2] + b)
// Wave per row, shuffle reduction.
// ---------------------------------------------------------------------------
__global__ void gemm_out2_relu_kernel(const float* __restrict__ H,
                                      const float* __restrict__ W,
                                      const float* __restrict__ b,
                                      float* __restrict__ out2) {
    int gid  = blockIdx.x * blockDim.x + threadIdx.x;
    int row  = gid >> 5;
    int lane = gid & 31;
    if (row >= N_NODES) return;
    float4 v = *(const float4*)(H + (long long)row * DIM + lane * 4);
    const float* wp = W + lane * 4 * 2;    // W row-major [128][2]
    float d0 = v.x * wp[0] + v.y * wp[2] + v.z * wp[4] + v.w * wp[6];
    float d1 = v.x * wp[1] + v.y * wp[3] + v.z * wp[5] + v.w * wp[7];
    for (int off = 16; off > 0; off >>= 1) {
        d0 += __shfl_down(d0, off, 32);
        d1 += __shfl_down(d1, off, 32);
    }
    if (lane == 0) {
        float o0 = d0 + b[0]; o0 = o0 > 0.0f ? o0 : 0.0f;
        float o1 = d1 + b[1]; o1 = o1 > 0.0f ? o1 : 0.0f;
        out2[row * 2 + 0] = o0;
        out2[row * 2 + 1] = o1;
    }
}

// ---------------------------------------------------------------------------
// 2-channel BatchNorm
// ---------------------------------------------------------------------------
__global__ void bn_reduce2_kernel(const float* __restrict__ h2,
                                  float* __restrict__ stats2) {
    int tid = blockIdx.x * blockDim.x + threadIdx.x;
    int stride = gridDim.x * blockDim.x;
    float s0 = 0.0f, s1 = 0.0f, q0 = 0.0f, q1 = 0.0f;
    for (int row = tid; row < N_NODES; row += stride) {
        float a = h2[row * 2 + 0];
        float c = h2[row * 2 + 1];
        s0 += a; q0 += a * a;
        s1 += c; q1 += c * c;
    }
    atomicAdd(&stats2[0], s0);
    atomicAdd(&stats2[1], s1);
    atomicAdd(&stats2[2], q0);
    atomicAdd(&stats2[3], q1);
}

__global__ void bn_apply2_kernel(const float* __restrict__ in2,
                                 const float* __restrict__ stats2,
                                 const float* __restrict__ g,
                                 const float* __restrict__ be,
                                 float* __restrict__ out) {
    const float invN = 1.0f / (float)N_NODES;
    int i = blockIdx.x * blockDim.x + threadIdx.x;
    const int total = N_NODES * 2;
    const int stride = gridDim.x * blockDim.x;
    for (; i < total; i += stride) {
        int c = i & 1;
        float mean = stats2[c] * invN;
        float var  = stats2[2 + c] * invN - mean * mean;
        float rs   = __frsqrt_rn(var + BN_EPS);
        out[i] = (in2[i] - mean) * rs * g[c] + be[c];
    }
}

// ---------------------------------------------------------------------------
// Host orchestration
// ---------------------------------------------------------------------------
extern "C" void kernel_launch(void* const* d_in, const int* in_sizes, int n_in,
                              void* d_out, int out_size, void* d_ws, size_t ws_size,
                              hipStream_t stream) {
    (void)in_sizes; (void)n_in; (void)out_size; (void)ws_size;

    const float*      x   = (const float*)d_in[0];
    const long long*  ei  = (const long long*)d_in[1];   // int64 edge_index [2][E]
    const float*      W1a = (const float*)d_in[2];
    const float*      b1a = (const float*)d_in[3];
    const float*      W1b = (const float*)d_in[4];
    const float*      b1b = (const float*)d_in[5];
    const float*      g1  = (const float*)d_in[6];
    const float*      be1 = (const float*)d_in[7];
    const float*      W2a = (const float*)d_in[8];
    const float*      b2a = (const float*)d_in[9];
    const float*      W2b = (const float*)d_in[10];
    const float*      b2b = (const float*)d_in[11];
    const float*      g2  = (const float*)d_in[12];
    const float*      be2 = (const float*)d_in[13];
    const float*      W5a = (const float*)d_in[14];
    const float*      b5a = (const float*)d_in[15];
    const float*      W5b = (const float*)d_in[16];
    const float*      b5b = (const float*)d_in[17];
    const float*      g5  = (const float*)d_in[18];
    const float*      be5 = (const float*)d_in[19];

    const long long* srcI = ei;            // edge_index[0]
    const long long* dstI = ei + N_EDGES;  // edge_index[1]

    // Workspace layout (floats)
    const size_t TENS = (size_t)N_NODES * DIM;
    float* bufA  = (float*)d_ws;                  // N x 128
    float* bufB  = bufA + TENS;                   // N x 128
    float* bufC  = bufB + TENS;                   // N x 128
    float* bufD  = bufC + TENS;                   // N x 2
    float* stats = bufD + (size_t)N_NODES * 2;    // 256 + 4 floats

    float* out = (float*)d_out;

    const int n4 = N_NODES * DIM / 4;               // float4 count
    const dim3 blk(256);
    const dim3 gCopy(4096);
    const dim3 gEdge((N_EDGES * 32 + 255) / 256);   // 75000
    const dim3 gGemm(N_NODES / 16 / TILES_PER_BLOCK); // 625 blocks x 5 tiles
    const dim3 gBnR(256);
    const dim3 gElem(4096);
    const dim3 gRow((N_NODES * 32 + 255) / 256);
    const dim3 gSmall(64);

    // ---------------- Layer 1 ----------------
    copy_vec4_kernel<<<gCopy, blk, 0, stream>>>((const float4*)x, (float4*)bufA, n4);
    edge_aggregate_kernel<<<gEdge, blk, 0, stream>>>(x, srcI, dstI, bufA);
    gemm128_relu_wmma_kernel<<<gGemm, blk, 0, stream>>>(bufA, W1a, b1a, bufB);
    gemm128_relu_wmma_kernel<<<gGemm, blk, 0, stream>>>(bufB, W1b, b1b, bufC);
    zero_kernel<<<dim3(2), blk, 0, stream>>>(stats, 260);
    bn_reduce128_kernel<<<gBnR, blk, 0, stream>>>(bufC, stats);
    bn_apply128_kernel<<<gElem, blk, 0, stream>>>(bufC, stats, g1, be1, bufA);

    // ---------------- Layer 2 ----------------
    copy_vec4_kernel<<<gCopy, blk, 0, stream>>>((const float4*)bufA, (float4*)bufB, n4);
    edge_aggregate_kernel<<<gEdge, blk, 0, stream>>>(bufA, srcI, dstI, bufB);
    gemm128_relu_wmma_kernel<<<gGemm, blk, 0, stream>>>(bufB, W2a, b2a, bufC);
    gemm128_relu_wmma_kernel<<<gGemm, blk, 0, stream>>>(bufC, W2b, b2b, bufB);
    zero_kernel<<<dim3(2), blk, 0, stream>>>(stats, 260);
    bn_reduce128_kernel<<<gBnR, blk, 0, stream>>>(bufB, stats);
    bn_apply128_kernel<<<gElem, blk, 0, stream>>>(bufB, stats, g2, be2, bufA);

    // ---------------- Layer 3 ----------------
    copy_vec4_kernel<<<gCopy, blk, 0, stream>>>((const float4*)bufA, (float4*)bufB, n4);
    edge_aggregate_kernel<<<gEdge, blk, 0, stream>>>(bufA, srcI, dstI, bufB);
    gemm128_relu_wmma_kernel<<<gGemm, blk, 0, stream>>>(bufB, W5a, b5a, bufC);
    gemm_out2_relu_kernel<<<gRow, blk, 0, stream>>>(bufC, W5b, b5b, bufD);
    zero_kernel<<<dim3(2), blk, 0, stream>>>(stats, 260);
    bn_reduce2_kernel<<<gSmall, blk, 0, stream>>>(bufD, stats + 256);
    bn_apply2_kernel<<<gSmall, blk, 0, stream>>>(bufD, stats + 256, g5, be5, out);
}